// dual_array_gen_feature_2877628088735
// MI455X (gfx1250) — compile-verified
//
#include <hip/hip_runtime.h>

typedef __attribute__((ext_vector_type(2))) float v2f;
typedef __attribute__((ext_vector_type(4))) float v4f;
typedef __attribute__((ext_vector_type(8))) float v8f;

#define H_IMG 256
#define W_IMG 256
#define NPIX  (H_IMG * W_IMG)   // 65536
#define COUT  96
#define KDIM  27                // 3 in-channels * 3 * 3
#define NPTS  2000000
#define RES   NPIX              // 65536 rows in the flat (res, 96) view

// im2col element X[k][pixel(h,w)] with zero 'SAME' padding; k >= 27 -> 0 pad of K
__device__ __forceinline__ float im2col_val(const float* __restrict__ img,
                                            int k, int h, int w) {
    if (k >= KDIM) return 0.0f;
    int i  = k / 9;
    int r  = k - i * 9;
    int kh = r / 3;
    int kw = r - kh * 3;
    int ih = h + kh - 1;
    int iw = w + kw - 1;
    if ((unsigned)ih >= (unsigned)H_IMG || (unsigned)iw >= (unsigned)W_IMG) return 0.0f;
    return img[i * NPIX + ih * W_IMG + iw];
}

// Conv as GEMM: feat[c, p] = sum_k W[c,k] * X[k,p] + bias[c]
// One wave computes one 16(M=channels) x 16(N=pixels) tile via 7x V_WMMA_F32_16X16X4_F32.
__global__ void __launch_bounds__(256) conv_wmma_kernel(
    const float* __restrict__ image,    // [3,256,256]
    const float* __restrict__ conv_w,   // [96,27] (OIHW flattened)
    const float* __restrict__ conv_b,   // [96]
    float* __restrict__ feat)           // [96,65536] in d_ws
{
    const int wave = (blockIdx.x * blockDim.x + threadIdx.x) >> 5;
    const int lane = threadIdx.x & 31;

    const int n_ntiles = NPIX / 16;          // 4096
    const int ctile = wave / n_ntiles;       // 0..5  (96/16)
    const int ntile = wave - ctile * n_ntiles;

    const int half = lane >> 4;              // 0: lanes 0-15, 1: lanes 16-31
    const int l15  = lane & 15;

    const int m = ctile * 16 + l15;          // A-matrix row (channel) for this lane
    const int n = ntile * 16 + l15;          // B-matrix column (pixel) for this lane
    const int h = n >> 8;
    const int w = n & 255;

    v8f acc = {};
    #pragma unroll
    for (int kstep = 0; kstep < 7; ++kstep) {
        // A layout 16x4 f32: lanes 0-15 hold K = {4k, 4k+1}, lanes 16-31 hold K = {4k+2, 4k+3}
        // B layout 4x16 f32: same K pairing per lane-half, N striped across lanes 0-15.
        const int kbase = kstep * 4 + half * 2;
        v2f a, b;
        a.x = (kbase     < KDIM) ? conv_w[m * KDIM + kbase]     : 0.0f;
        a.y = (kbase + 1 < KDIM) ? conv_w[m * KDIM + kbase + 1] : 0.0f;
        b.x = im2col_val(image, kbase,     h, w);
        b.y = im2col_val(image, kbase + 1, h, w);
        acc = __builtin_amdgcn_wmma_f32_16x16x4_f32(
            /*neg_a=*/false, a, /*neg_b=*/false, b,
            /*c_mod=*/(short)0, acc, /*reuse_a=*/false, /*reuse_b=*/false);
    }

    // D layout 16x16 f32: VGPR r -> M = r + 8*half, N = lane&15
    #pragma unroll
    for (int r = 0; r < 8; ++r) {
        const int mm = ctile * 16 + r + half * 8;
        feat[mm * NPIX + n] = acc[r] + conv_b[mm];
    }
}

// Gather + 1D-bilinear blend. 12 threads (x-dim) per point, each handles one
// float4 chunk of the 48-float output row -> coalesced 192B bursts for all
// four gathered rows and for the output store. Feature table (24 MB) stays
// L2-resident; output stream (384 MB) is stored non-temporally so it does not
// evict it.
__global__ void __launch_bounds__(192) gather_interp_kernel(
    const v2f* __restrict__ pts,   // [NPTS] (x,y)
    const v4f* __restrict__ fa4,   // feat viewed as [65536][24] float4
    v4f* __restrict__ out4)        // [NPTS][12] float4
{
    const int p     = blockIdx.x * 16 + threadIdx.y;  // point index
    const int chunk = threadIdx.x;                    // 0..11

    const v2f pt = pts[p];
    const float scale = (float)(RES - 1);                       // 65535
    const float hi    = (float)((double)(RES - 1) - 1e-5);      // == 65535.0f in f32

    float x = pt.x * scale;
    float y = pt.y * scale;
    x = fminf(fmaxf(x, 0.0f), hi);
    y = fminf(fmaxf(y, 0.0f), hi);

    const int x1 = (int)floorf(x);
    const int y1 = (int)floorf(y);
    const int x2 = min(x1 + 1, RES - 1);
    const int y2 = min(y1 + 1, RES - 1);

    const float xw1 = (float)x2 - x;
    const float xw2 = x - (float)x1;
    const float yw1 = (float)y2 - y;
    const float yw2 = y - (float)y1;

    // fx = cols [0,48)  -> float4 chunks [0,12); fy = cols [48,96) -> chunks [12,24)
    const v4f fx1 = fa4[x1 * 24 + chunk];
    const v4f fx2 = fa4[x2 * 24 + chunk];
    const v4f fy1 = fa4[y1 * 24 + 12 + chunk];
    const v4f fy2 = fa4[y2 * 24 + 12 + chunk];

    v4f o;
    o.x = xw1 * fx1.x + xw2 * fx2.x + yw1 * fy1.x + yw2 * fy2.x;
    o.y = xw1 * fx1.y + xw2 * fx2.y + yw1 * fy1.y + yw2 * fy2.y;
    o.z = xw1 * fx1.z + xw2 * fx2.z + yw1 * fy1.z + yw2 * fy2.z;
    o.w = xw1 * fx1.w + xw2 * fx2.w + yw1 * fy1.w + yw2 * fy2.w;

    __builtin_nontemporal_store(o, &out4[p * 12 + chunk]);
}

extern "C" void kernel_launch(void* const* d_in, const int* in_sizes, int n_in,
                              void* d_out, int out_size, void* d_ws, size_t ws_size,
                              hipStream_t stream) {
    const float* pts    = (const float*)d_in[0];   // [2M, 2]
    const float* image  = (const float*)d_in[1];   // [1, 3, 256, 256]
    const float* conv_w = (const float*)d_in[2];   // [96, 3, 3, 3]
    const float* conv_b = (const float*)d_in[3];   // [96]

    float* feat = (float*)d_ws;                    // [96, 65536] = 24 MB scratch

    // 6 ctiles * 4096 ntiles = 24576 waves; 8 waves (256 threads) per block.
    conv_wmma_kernel<<<24576 / 8, 256, 0, stream>>>(image, conv_w, conv_b, feat);

    // 2M points, 16 points per block, 12 chunk-threads per point.
    dim3 gblock(12, 16, 1);
    gather_interp_kernel<<<NPTS / 16, gblock, 0, stream>>>(
        (const v2f*)pts, (const v4f*)feat, (v4f*)d_out);
}